// GAT_82575041232959
// MI455X (gfx1250) — compile-verified
//
#include <hip/hip_runtime.h>
#include <hip/hip_bf16.h>
#include <math.h>

#define NEG_SLOPE 0.2f
#define BN_EPS 1e-5f

typedef __attribute__((ext_vector_type(2))) float v2f;
typedef __attribute__((ext_vector_type(8))) float v8f;

// ---------------------------------------------------------------------------
// GEMM: C[n,Ko] = A[n,Ki] * W[Ki,Ko], all f32, via V_WMMA_F32_16X16X4_F32.
// One 16x16 output tile per wave32. K-step = 4.
// A frag (16x4, 2 VGPR): lanes 0-15 -> M=lane,  K = {k, k+1}
//                        lanes16-31 -> M=lane-16, K = {k+2, k+3}
// B frag (4x16, 2 VGPR): lanes 0-15 -> N=lane,  K = {k, k+1}
//                        lanes16-31 -> N=lane-16, K = {k+2, k+3}
// C/D  (16x16, 8 VGPR):  vgpr r, lanes 0-15 -> (M=r,   N=lane)
//                                lanes16-31 -> (M=r+8, N=lane-16)
// ---------------------------------------------------------------------------
__global__ __launch_bounds__(128) void gemm_f32_wmma(
    const float* __restrict__ A, const float* __restrict__ W,
    float* __restrict__ C, int n, int Ki, int Ko)
{
  const int lane = threadIdx.x & 31;
  const int wave = threadIdx.x >> 5;
  const int colTiles = Ko >> 4;
  const int tile = blockIdx.x * 4 + wave;
  const int rowTile = tile / colTiles;
  const int colTile = tile % colTiles;
  if (rowTile * 16 >= n) return;
  const int row0 = rowTile * 16;
  const int col0 = colTile * 16;
  const int half = lane >> 4;
  const int mr = lane & 15;   // M index for A, N index for B

  const float* __restrict__ arow = A + (size_t)(row0 + mr) * Ki + 2 * half;
  const float* __restrict__ bcol = W + (size_t)(2 * half) * Ko + col0 + mr;

  v8f acc = {};
#pragma unroll 4
  for (int k = 0; k < Ki; k += 4) {
    v2f a, b;
    a.x = arow[k];
    a.y = arow[k + 1];
    b.x = bcol[(size_t)k * Ko];
    b.y = bcol[(size_t)k * Ko + Ko];
    acc = __builtin_amdgcn_wmma_f32_16x16x4_f32(
        false, a, false, b, (short)0, acc, false, false);
  }
#pragma unroll
  for (int r = 0; r < 8; ++r) {
    const int row = row0 + r + 8 * half;
    C[(size_t)row * Ko + col0 + mr] = acc[r];
  }
}

// ---------------------------------------------------------------------------
// Per-(node, head) attention logits: als = <h, a_s>, ald = <h, a_d>
// ---------------------------------------------------------------------------
__global__ void attn_logits_kernel(
    const float* __restrict__ h, const float* __restrict__ a_s,
    const float* __restrict__ a_d, float* __restrict__ als,
    float* __restrict__ ald, int n, int heads, int ch)
{
  const int i = blockIdx.x * blockDim.x + threadIdx.x;
  if (i >= n * heads) return;
  const int node = i / heads, hd = i % heads;
  const float* __restrict__ hp = h + (size_t)node * heads * ch + hd * ch;
  const float* __restrict__ as = a_s + hd * ch;
  const float* __restrict__ ad = a_d + hd * ch;
  float s = 0.f, d = 0.f;
  for (int c = 0; c < ch; ++c) {
    const float v = hp[c];
    s = fmaf(v, as[c], s);
    d = fmaf(v, ad[c], d);
  }
  als[i] = s;
  ald[i] = d;
}

// Order-preserving float <-> uint encoding so segment-max can use the fast
// hardware u32 atomic-max (global_atomic_max_u32).
__device__ __forceinline__ unsigned f2ord(float f) {
  const unsigned u = __float_as_uint(f);
  return (u >> 31) ? ~u : (u | 0x80000000u);
}
__device__ __forceinline__ float ord2f(unsigned e) {
  return (e >> 31) ? __uint_as_float(e ^ 0x80000000u) : __uint_as_float(~e);
}

__device__ __forceinline__ void edge_pair(const int* __restrict__ ei, int e,
                                          int E, int& s, int& d) {
  if (e < E) { s = ei[e]; d = ei[E + e]; }   // edge_index[0][e], [1][e]
  else       { s = d = e - E; }              // self loops
}

// ---------------------------------------------------------------------------
// Pass 1: segment max of leaky_relu(als[src] + ald[dst]) into m[dst]
// ---------------------------------------------------------------------------
__global__ void edge_max_kernel(
    const int* __restrict__ ei, int E, int N,
    const float* __restrict__ als, const float* __restrict__ ald,
    unsigned* __restrict__ m, int heads)
{
  const int e = blockIdx.x * blockDim.x + threadIdx.x;
  if (e >= E + N) return;
  int s, d; edge_pair(ei, e, E, s, d);
  for (int hd = 0; hd < heads; ++hd) {
    float v = als[s * heads + hd] + ald[d * heads + hd];
    v = v > 0.f ? v : NEG_SLOPE * v;
    atomicMax(&m[d * heads + hd], f2ord(v));
  }
}

// ---------------------------------------------------------------------------
// Pass 2: denom[dst] += exp(e - m[dst])
// ---------------------------------------------------------------------------
__global__ void edge_sum_kernel(
    const int* __restrict__ ei, int E, int N,
    const float* __restrict__ als, const float* __restrict__ ald,
    const unsigned* __restrict__ m, float* __restrict__ denom, int heads)
{
  const int e = blockIdx.x * blockDim.x + threadIdx.x;
  if (e >= E + N) return;
  int s, d; edge_pair(ei, e, E, s, d);
  for (int hd = 0; hd < heads; ++hd) {
    float v = als[s * heads + hd] + ald[d * heads + hd];
    v = v > 0.f ? v : NEG_SLOPE * v;
    atomicAdd(&denom[d * heads + hd], __expf(v - ord2f(m[d * heads + hd])));
  }
}

// ---------------------------------------------------------------------------
// Pass 3: out[dst] += h[src] * alpha.  One wave per edge; 32 lanes stride
// the D = heads*ch channels. Alphas precomputed per head (heads <= 4).
// ---------------------------------------------------------------------------
__global__ void edge_agg_kernel(
    const int* __restrict__ ei, int E, int N, const float* __restrict__ h,
    const float* __restrict__ als, const float* __restrict__ ald,
    const unsigned* __restrict__ m, const float* __restrict__ denom,
    float* __restrict__ out, int heads, int ch)
{
  const int lane = threadIdx.x & 31;
  const int e = (blockIdx.x * blockDim.x + threadIdx.x) >> 5;
  if (e >= E + N) return;
  int s, d; edge_pair(ei, e, E, s, d);
  const int D = heads * ch;
  float alpha[4];
  for (int hd = 0; hd < heads; ++hd) {
    float v = als[s * heads + hd] + ald[d * heads + hd];
    v = v > 0.f ? v : NEG_SLOPE * v;
    alpha[hd] = __expf(v - ord2f(m[d * heads + hd])) /
                (denom[d * heads + hd] + 1e-16f);
  }
  const float* __restrict__ hs = h + (size_t)s * D;
  float* __restrict__ od = out + (size_t)d * D;
  for (int c = lane; c < D; c += 32)
    atomicAdd(&od[c], hs[c] * alpha[c / ch]);
}

// ---------------------------------------------------------------------------
// BN reduce: per-channel sum & sum-of-squares of (x + bias). blockDim == D.
// ---------------------------------------------------------------------------
__global__ void bn_reduce_kernel(
    const float* __restrict__ x, const float* __restrict__ bias,
    float* __restrict__ stats, int n, int D)
{
  const int c = threadIdx.x;
  const float bc = bias[c];
  float s = 0.f, sq = 0.f;
  for (int r = blockIdx.x; r < n; r += gridDim.x) {
    const float v = x[(size_t)r * D + c] + bc;
    s += v;
    sq = fmaf(v, v, sq);
  }
  atomicAdd(&stats[c], s);
  atomicAdd(&stats[D + c], sq);
}

// ---------------------------------------------------------------------------
// Fused bias + batchnorm + ELU
// ---------------------------------------------------------------------------
__global__ void bn_elu_kernel(
    const float* __restrict__ x, const float* __restrict__ bias,
    const float* __restrict__ stats, const float* __restrict__ g,
    const float* __restrict__ be, float* __restrict__ y, int n, int D)
{
  const int i = blockIdx.x * blockDim.x + threadIdx.x;
  if (i >= n * D) return;
  const int c = i % D;
  const float inv = 1.0f / (float)n;
  const float mu = stats[c] * inv;
  const float var = fmaxf(stats[D + c] * inv - mu * mu, 0.f);
  float v = (x[i] + bias[c] - mu) * rsqrtf(var + BN_EPS) * g[c] + be[c];
  y[i] = v > 0.f ? v : expm1f(v);
}

// ---------------------------------------------------------------------------
// Host-side: one GAT layer (GEMM -> logits -> 3 edge passes -> BN+ELU)
// ---------------------------------------------------------------------------
static void gat_layer(const float* in, const float* W, const float* a_s,
                      const float* a_d, const float* bias, const float* g,
                      const float* be, float* hbuf, float* aggbuf,
                      float* outbuf, float* als, float* ald, unsigned* mbuf,
                      float* denom, float* stats, const int* ei, int N, int E,
                      int Ki, int heads, int ch, hipStream_t stream)
{
  const int D = heads * ch;
  const int tiles = (N / 16) * (D / 16);
  gemm_f32_wmma<<<(tiles + 3) / 4, 128, 0, stream>>>(in, W, hbuf, N, Ki, D);

  const int nh = N * heads;
  attn_logits_kernel<<<(nh + 255) / 256, 256, 0, stream>>>(hbuf, a_s, a_d, als,
                                                           ald, N, heads, ch);

  hipMemsetAsync(mbuf, 0, (size_t)N * heads * 4, stream);   // 0 == ord(-huge)
  hipMemsetAsync(denom, 0, (size_t)N * heads * 4, stream);
  hipMemsetAsync(aggbuf, 0, (size_t)N * D * 4, stream);
  hipMemsetAsync(stats, 0, (size_t)2 * D * 4, stream);

  const int tot = E + N;
  edge_max_kernel<<<(tot + 255) / 256, 256, 0, stream>>>(ei, E, N, als, ald,
                                                         mbuf, heads);
  edge_sum_kernel<<<(tot + 255) / 256, 256, 0, stream>>>(ei, E, N, als, ald,
                                                         mbuf, denom, heads);
  edge_agg_kernel<<<(tot + 7) / 8, 256, 0, stream>>>(ei, E, N, hbuf, als, ald,
                                                     mbuf, denom, aggbuf,
                                                     heads, ch);

  bn_reduce_kernel<<<512, D, 0, stream>>>(aggbuf, bias, stats, N, D);
  const int nd = N * D;
  bn_elu_kernel<<<(nd + 255) / 256, 256, 0, stream>>>(aggbuf, bias, stats, g,
                                                      be, outbuf, N, D);
}

extern "C" void kernel_launch(void* const* d_in, const int* in_sizes, int n_in,
                              void* d_out, int out_size, void* d_ws,
                              size_t ws_size, hipStream_t stream)
{
  const float* x   = (const float*)d_in[0];
  const int*   ei  = (const int*)d_in[1];   // int32 (JAX x64 disabled)
  const float* W1  = (const float*)d_in[2];
  const float* as1 = (const float*)d_in[3];
  const float* ad1 = (const float*)d_in[4];
  const float* b1  = (const float*)d_in[5];
  const float* g1  = (const float*)d_in[6];
  const float* be1 = (const float*)d_in[7];
  const float* W2  = (const float*)d_in[8];
  const float* as2 = (const float*)d_in[9];
  const float* ad2 = (const float*)d_in[10];
  const float* b2  = (const float*)d_in[11];
  const float* g2  = (const float*)d_in[12];
  const float* be2 = (const float*)d_in[13];
  const float* W3  = (const float*)d_in[14];
  const float* as3 = (const float*)d_in[15];
  const float* ad3 = (const float*)d_in[16];
  const float* b3  = (const float*)d_in[17];
  const float* g3  = (const float*)d_in[18];
  const float* be3 = (const float*)d_in[19];

  const int N = in_sizes[0] / 128;   // 50000
  const int E = in_sizes[1] / 2;     // 800000

  // Workspace layout (floats): P[N*128] Q[N*128] als[N*4] ald[N*4]
  //                            m[N*4](u32) denom[N*4] stats[256]
  float*    P     = (float*)d_ws;
  float*    Q     = P + (size_t)N * 128;
  float*    als   = Q + (size_t)N * 128;
  float*    ald   = als + (size_t)N * 4;
  unsigned* mbuf  = (unsigned*)(ald + (size_t)N * 4);
  float*    denom = (float*)(mbuf + (size_t)N * 4);
  float*    stats = denom + (size_t)N * 4;

  // Layer 1: x -> h(P), agg(Q), out(P)
  gat_layer(x, W1, as1, ad1, b1, g1, be1, P, Q, P, als, ald, mbuf, denom,
            stats, ei, N, E, 128, 4, 32, stream);
  // Layer 2: P -> h(Q), agg(P), out(Q)
  gat_layer(P, W2, as2, ad2, b2, g2, be2, Q, P, Q, als, ald, mbuf, denom,
            stats, ei, N, E, 128, 4, 32, stream);
  // Layer 3: Q -> h(P[0:N*64]), agg(P[N*64:N*128]), out(d_out); heads=1, ch=64
  float* h3   = P;
  float* agg3 = P + (size_t)N * 64;
  gat_layer(Q, W3, as3, ad3, b3, g3, be3, h3, agg3, (float*)d_out, als, ald,
            mbuf, denom, stats, ei, N, E, 128, 1, 64, stream);
}